// GazeLoss_58059367908111
// MI455X (gfx1250) — compile-verified
//
#include <hip/hip_runtime.h>

typedef __attribute__((ext_vector_type(2))) float v2f;
typedef __attribute__((ext_vector_type(8))) float v8f;

#define BATCH   64
#define CHANS   3
#define IMG_H   512
#define IMG_W   512
#define PATCH   32
#define PAD_F   0.3f
// total elements per eye-mean: B*C*PATCH*PATCH = 196608; final scale = 1/(2*196608)
#define INV_2N  (1.0f / 393216.0f)

// Full-precision wave32 reduction through the matrix pipe:
// D = A(ones,16x4) * B(4x16 = per-lane partials) + 0  => every row of D holds
// the 16 column sums of B. Each lane contributes its accumulator once (b.x),
// so sum over all 32 lanes of D[0] = 2 * (wave total)  (lanes 16..31 carry
// row M=8, identical to row 0).
__device__ __forceinline__ float wave_reduce_f32_wmma(float acc) {
    v2f a; a.x = 1.0f; a.y = 1.0f;       // A-matrix: all ones
    v2f b; b.x = acc;  b.y = 0.0f;       // B-matrix: one live element per lane
    v8f c = {};
    v8f d = __builtin_amdgcn_wmma_f32_16x16x4_f32(
        /*neg_a=*/false, a, /*neg_b=*/false, b,
        /*c_mod=*/(short)0, c, /*reuse_a=*/false, /*reuse_b=*/false);
    float v = d[0];
    #pragma unroll
    for (int off = 16; off > 0; off >>= 1)
        v += __shfl_xor(v, off, 32);
    return v * 0.5f;
}

__device__ __forceinline__ float bilinear_fetch(const float* __restrict__ img,
                                                int y0, int y1, int x0, int x1,
                                                float wx, float wy) {
    float Ia = img[y0 * IMG_W + x0];
    float Ib = img[y0 * IMG_W + x1];
    float Ic = img[y1 * IMG_W + x0];
    float Id = img[y1 * IMG_W + x1];
    float top = Ia + wx * (Ib - Ia);
    float bot = Ic + wx * (Id - Ic);
    return top + wy * (bot - top);
}

// grid = (channel=3, eye=2, batch=64), block = 256 threads (8 waves),
// each thread does 4 of the 32x32 samples for pred & target.
__global__ void gaze_patch_l1_kernel(const float* __restrict__ pred,
                                     const float* __restrict__ target,
                                     const float* __restrict__ lm,
                                     float* __restrict__ partials) {
    const int c   = blockIdx.x;
    const int eye = blockIdx.y;
    const int b   = blockIdx.z;

    // --- eye bbox from 6 landmarks (uniform per block -> scalar loads) ---
    const float* p = lm + (size_t)b * (68 * 2) + (size_t)(36 + 6 * eye) * 2;
    float xmn = p[0], xmx = p[0], ymn = p[1], ymx = p[1];
    #pragma unroll
    for (int j = 1; j < 6; ++j) {
        float x = p[2 * j], y = p[2 * j + 1];
        xmn = fminf(xmn, x); xmx = fmaxf(xmx, x);
        ymn = fminf(ymn, y); ymx = fmaxf(ymx, y);
    }
    const float w = xmx - xmn, h = ymx - ymn;
    const float maxX = (float)(IMG_W - 1), maxY = (float)(IMG_H - 1);
    const float bx1 = fminf(fmaxf(xmn - w * PAD_F, 0.0f), maxX);
    const float by1 = fminf(fmaxf(ymn - h * PAD_F, 0.0f), maxY);
    const float bx2 = fminf(fmaxf(xmx + w * PAD_F, 0.0f), maxX);
    const float by2 = fminf(fmaxf(ymx + h * PAD_F, 0.0f), maxY);
    const bool degen = (bx2 - bx1 < 1.0f) || (by2 - by1 < 1.0f);
    // degenerate grid: gx=gy=0 -> px=py=0.5*(W-1)
    const float cpx = 0.5f * maxX, cpy = 0.5f * maxY;

    const size_t img_off = ((size_t)b * CHANS + (size_t)c) * (size_t)(IMG_H * IMG_W);
    const float* __restrict__ pim = pred + img_off;
    const float* __restrict__ tim = target + img_off;

    float acc = 0.0f;
    #pragma unroll
    for (int k = 0; k < 4; ++k) {
        const int s  = (int)threadIdx.x + k * 256;   // 0..1023
        const int iy = s >> 5;
        const int ix = s & 31;
        const float tx = (float)ix * (1.0f / (PATCH - 1));
        const float ty = (float)iy * (1.0f / (PATCH - 1));
        const float px = degen ? cpx : fminf(fmaxf(bx1 + (bx2 - bx1) * tx, 0.0f), maxX);
        const float py = degen ? cpy : fminf(fmaxf(by1 + (by2 - by1) * ty, 0.0f), maxY);

        const float fx = floorf(px), fy = floorf(py);
        const float wx = px - fx,  wy = py - fy;
        const int x0 = (int)fx;
        const int y0 = (int)fy;
        const int x1 = min(x0 + 1, IMG_W - 1);
        const int y1 = min(y0 + 1, IMG_H - 1);

        const float vp = bilinear_fetch(pim, y0, y1, x0, x1, wx, wy);
        const float vt = bilinear_fetch(tim, y0, y1, x0, x1, wx, wy);
        acc += fabsf(vp - vt);
    }

    // wave-level reduction via the f32 matrix pipe (EXEC all-ones: no divergence here)
    const float wsum = wave_reduce_f32_wmma(acc);

    __shared__ float lds[8];
    const int lane = threadIdx.x & 31;
    const int wid  = threadIdx.x >> 5;
    if (lane == 0) lds[wid] = wsum;
    __syncthreads();
    if (threadIdx.x == 0) {
        float t = 0.0f;
        #pragma unroll
        for (int i = 0; i < 8; ++i) t += lds[i];
        partials[((size_t)b * 2 + (size_t)eye) * CHANS + (size_t)c] = t;
    }
}

// single block reduces the 384 block partials deterministically
__global__ void gaze_final_reduce_kernel(const float* __restrict__ partials,
                                         float* __restrict__ out, int n) {
    float v = ((int)threadIdx.x < n) ? partials[threadIdx.x] : 0.0f;
    #pragma unroll
    for (int off = 16; off > 0; off >>= 1)
        v += __shfl_xor(v, off, 32);
    __shared__ float lds[12];
    const int lane = threadIdx.x & 31;
    const int wid  = threadIdx.x >> 5;
    if (lane == 0) lds[wid] = v;
    __syncthreads();
    if (threadIdx.x == 0) {
        float s = 0.0f;
        #pragma unroll
        for (int i = 0; i < 12; ++i) s += lds[i];
        out[0] = s * INV_2N;
    }
}

extern "C" void kernel_launch(void* const* d_in, const int* in_sizes, int n_in,
                              void* d_out, int out_size, void* d_ws, size_t ws_size,
                              hipStream_t stream) {
    const float* pred   = (const float*)d_in[0];
    const float* target = (const float*)d_in[1];
    const float* lm     = (const float*)d_in[2];
    float* out      = (float*)d_out;
    float* partials = (float*)d_ws;   // 384 floats

    dim3 grid(CHANS, 2, BATCH);       // 384 blocks
    gaze_patch_l1_kernel<<<grid, 256, 0, stream>>>(pred, target, lm, partials);
    gaze_final_reduce_kernel<<<1, 384, 0, stream>>>(partials, out, CHANS * 2 * BATCH);
}